// CostVolume1D_45767171506374
// MI455X (gfx1250) — compile-verified
//
#include <hip/hip_runtime.h>

// Cost volume 1D (9-shift correlation, channel mean) on gfx1250 via
// V_WMMA_F32_16X16X4_F32. One wave32 per 16-wide output tile:
//   acc0(16x16) = f1[16 w] x f2[w0-4 .. w0+11]   (K = C = 128, 32 WMMAs)
//   acc1(16x16) = f1[16 w] x f2[w0+12 .. w0+27]  (only 8 cols used)
// Diagonals i = n - m (acc0) / i = n - m + 16 (acc1) give all 9 shifts.

typedef __attribute__((ext_vector_type(2))) float v2f;
typedef __attribute__((ext_vector_type(8))) float v8f;

#define NB   4
#define CC   128
#define HH   192
#define WW   640
#define DD   4
#define SH   9            // 2*DD + 1 shifts
#define HWS  (HH * WW)    // channel stride

__global__ __launch_bounds__(128) void cost_volume_wmma(
    const float* __restrict__ f1, const float* __restrict__ f2,
    float* __restrict__ out)
{
    const int lane = threadIdx.x & 31;
    const int wv   = threadIdx.x >> 5;
    const int gw   = blockIdx.x * 4 + wv;     // global tile id
    const int tilesW = WW / 16;               // 40
    const int tw = gw % tilesW;
    const int t  = gw / tilesW;
    const int h  = t % HH;
    const int n  = t / HH;
    const int w0 = tw * 16;

    const int lm = lane & 15;                 // A: row M / B: col N / C,D: col N
    const int hi = lane >> 4;                 // selects K pair (A,B) / M half (C,D)

    // Base offsets at channel 0.
    const size_t nBase = (size_t)n * CC * HWS + (size_t)h * WW;

    // A fragment: f1[n, c, h, w0+lm]; lanes 0-15 K={0,1}, lanes 16-31 K={2,3}.
    const float* pA = f1 + nBase + (size_t)(w0 + lm) + (size_t)(2 * hi) * HWS;

    // B fragments: f2 columns, zero-padded outside [0, WW).
    const int col0 = w0 - DD + lm;            // acc0 column
    const int col1 = w0 + 12 + lm;            // acc1 column
    const bool ok0 = (col0 >= 0) & (col0 < WW);
    const bool ok1 = (col1 < WW);
    const int c0c = ok0 ? col0 : 0;           // clamped (avoid OOB loads)
    const int c1c = ok1 ? col1 : 0;
    const float* pB0 = f2 + nBase + (size_t)c0c + (size_t)(2 * hi) * HWS;
    const float* pB1 = f2 + nBase + (size_t)c1c + (size_t)(2 * hi) * HWS;

    v8f acc0 = {};
    v8f acc1 = {};

#pragma unroll 8
    for (int k = 0; k < CC; k += 4) {
        v2f A, B0, B1;
        A.x = pA[0];
        A.y = pA[HWS];
        const float t00 = pB0[0], t01 = pB0[HWS];
        const float t10 = pB1[0], t11 = pB1[HWS];
        B0.x = ok0 ? t00 : 0.0f;  B0.y = ok0 ? t01 : 0.0f;
        B1.x = ok1 ? t10 : 0.0f;  B1.y = ok1 ? t11 : 0.0f;

        acc0 = __builtin_amdgcn_wmma_f32_16x16x4_f32(
            false, A, false, B0, (short)0, acc0, false, false);
        acc1 = __builtin_amdgcn_wmma_f32_16x16x4_f32(
            false, A, false, B1, (short)0, acc1, false, false);

        pA  += 4 * (size_t)HWS;
        pB0 += 4 * (size_t)HWS;
        pB1 += 4 * (size_t)HWS;
    }

    // C/D layout: VGPR r holds (M = r + 8*hi, N = lm).
    const float scale = 1.0f / (float)CC;
#pragma unroll
    for (int r = 0; r < 8; ++r) {
        const int M  = r + 8 * hi;
        const int wg = w0 + M;
        const int i0 = lm - M;                // shift index from acc0 (f2 col w0-4+lm)
        if (i0 >= 0 && i0 < SH)
            out[(((size_t)n * SH + i0) * HH + h) * WW + wg] = acc0[r] * scale;
        const int i1 = 16 + lm - M;           // shift index from acc1 (f2 col w0+12+lm)
        if (i1 >= 0 && i1 < SH)
            out[(((size_t)n * SH + i1) * HH + h) * WW + wg] = acc1[r] * scale;
    }
}

extern "C" void kernel_launch(void* const* d_in, const int* in_sizes, int n_in,
                              void* d_out, int out_size, void* d_ws, size_t ws_size,
                              hipStream_t stream) {
    const float* f1 = (const float*)d_in[0];
    const float* f2 = (const float*)d_in[1];
    float* out = (float*)d_out;

    const int tiles  = NB * HH * (WW / 16);   // 30720 wave-tiles
    const int blocks = tiles / 4;             // 4 waves per 128-thread block
    cost_volume_wmma<<<blocks, 128, 0, stream>>>(f1, f2, out);
}